// GINNet_24129126269180
// MI455X (gfx1250) — compile-verified
//
#include <hip/hip_runtime.h>
#include <hip/hip_bf16.h>

#define GN_NODES   25000
#define GN_EDGES   400000
#define GN_D       128
#define GN_LAYERS  3
#define GN_GRAPHS  512
#define GN_DOUT    10
#define GN_BNEPS   1e-5f
#define GN_MTILES  ((GN_NODES + 15) / 16)   // 1563

typedef __attribute__((ext_vector_type(2))) float v2f;
typedef __attribute__((ext_vector_type(8))) float v8f;

// ---------------------------------------------------------------------------
// agg = (1 + eps[layer]) * h          (elementwise over N_NODES*D)
// ---------------------------------------------------------------------------
__global__ __launch_bounds__(256)
void gin_init_agg(const float* __restrict__ h, float* __restrict__ agg,
                  const float* __restrict__ epsP, int layer, int n) {
    int i = blockIdx.x * blockDim.x + threadIdx.x;
    if (i < n) agg[i] = (1.0f + epsP[layer]) * h[i];
}

// ---------------------------------------------------------------------------
// agg[dst[e]] += h[src[e]]            (one wave32 per edge, float4 per lane)
// ---------------------------------------------------------------------------
__global__ __launch_bounds__(256)
void gin_edge_agg(const float* __restrict__ h, const int* __restrict__ src,
                  const int* __restrict__ dst, float* __restrict__ agg,
                  int nEdges) {
    int t = blockIdx.x * blockDim.x + threadIdx.x;
    int e = t >> 5;
    int lane = t & 31;
    if (e >= nEdges) return;
    int s = src[e];
    int d = dst[e];
    const float4* hs = (const float4*)(h + (size_t)s * GN_D);
    float4 v = hs[lane];                       // 32 lanes * 4 floats = 128
    float* ad = agg + (size_t)d * GN_D + lane * 4;
    atomicAdd(ad + 0, v.x);
    atomicAdd(ad + 1, v.y);
    atomicAdd(ad + 2, v.z);
    atomicAdd(ad + 3, v.w);
}

// ---------------------------------------------------------------------------
// Out = epilogue(A @ W + bias)   A:[M x 128], W:[128 x 128] row-major
// One wave computes a 16(M) x 128(N) strip: A fragment loaded once per
// K-step and reused by 8 fp32 WMMAs (16x16x4). Branchless M-edge handling.
// BN=true : z = (z - mean)*scale + shift ; relu      (eval-mode batchnorm)
// BN=false: relu(z + bias)
// ---------------------------------------------------------------------------
template <bool BN>
__global__ __launch_bounds__(128)
void gin_gemm_epilogue(const float* __restrict__ A, const float* __restrict__ W,
                       const float* __restrict__ bias,
                       const float* __restrict__ gamma, const float* __restrict__ beta,
                       const float* __restrict__ mean,  const float* __restrict__ var,
                       float* __restrict__ Out, int M) {
    const int wave = threadIdx.x >> 5;                  // 4 waves / block
    const int tm   = blockIdx.x * 4 + wave;             // M tile
    if (tm >= GN_MTILES) return;                        // whole-wave exit

    const int lane = threadIdx.x & 31;
    const int half = lane >> 4;                         // K pair {0,1} vs {2,3}
    const int l    = lane & 15;

    const int row   = tm * 16 + l;                      // A row this lane feeds
    const bool rowok = (row < M);
    const float rmask = rowok ? 1.0f : 0.0f;            // branchless pad-zeroing
    const size_t arow = (size_t)(rowok ? row : (M - 1)) * GN_D;

    v8f acc[8];
    #pragma unroll
    for (int tn = 0; tn < 8; ++tn) acc[tn] = (v8f){};

    for (int kk = 0; kk < GN_D; kk += 4) {
        const int ka = kk + half * 2;
        const float2 av = *(const float2*)(A + arow + ka);
        v2f a;
        a[0] = av.x * rmask;
        a[1] = av.y * rmask;
        const float* w0 = W + (size_t)(ka + 0) * GN_D + l;
        const float* w1 = W + (size_t)(ka + 1) * GN_D + l;
        #pragma unroll
        for (int tn = 0; tn < 8; ++tn) {
            v2f b;
            b[0] = w0[tn * 16];
            b[1] = w1[tn * 16];
            acc[tn] = __builtin_amdgcn_wmma_f32_16x16x4_f32(
                          /*neg_a=*/false, a, /*neg_b=*/false, b,
                          /*c_mod=*/(short)0, acc[tn],
                          /*reuse_a=*/false, /*reuse_b=*/false);
        }
    }

    #pragma unroll
    for (int tn = 0; tn < 8; ++tn) {
        const int col = tn * 16 + l;
        float sc = 1.0f, sh;
        if (BN) {
            sc = gamma[col] * rsqrtf(var[col] + GN_BNEPS);
            sh = (bias[col] - mean[col]) * sc + beta[col];
        } else {
            sh = bias[col];
        }
        #pragma unroll
        for (int v = 0; v < 8; ++v) {
            const int r = tm * 16 + v + half * 8;       // C/D layout: VGPR v -> M row
            if (r < M) {
                float z = BN ? (acc[tn][v] * sc + sh) : (acc[tn][v] + sh);
                Out[(size_t)r * GN_D + col] = fmaxf(z, 0.0f);
            }
        }
    }
}

// ---------------------------------------------------------------------------
// readout
// ---------------------------------------------------------------------------
__global__ __launch_bounds__(256)
void gin_zero(float* __restrict__ p, int n) {
    int i = blockIdx.x * blockDim.x + threadIdx.x;
    if (i < n) p[i] = 0.0f;
}

__global__ __launch_bounds__(256)
void gin_pool(const float* __restrict__ h, const int* __restrict__ batch,
              float* __restrict__ sums, float* __restrict__ counts, int nNodes) {
    int t = blockIdx.x * blockDim.x + threadIdx.x;
    int node = t >> 5;
    int lane = t & 31;
    if (node >= nNodes) return;
    int g = batch[node];
    const float4* hr = (const float4*)(h + (size_t)node * GN_D);
    float4 v = hr[lane];
    float* s = sums + (size_t)g * GN_D + lane * 4;
    atomicAdd(s + 0, v.x);
    atomicAdd(s + 1, v.y);
    atomicAdd(s + 2, v.z);
    atomicAdd(s + 3, v.w);
    if (lane == 0) atomicAdd(counts + g, 1.0f);
}

__global__ __launch_bounds__(256)
void gin_out(const float* __restrict__ sums, const float* __restrict__ counts,
             const float* __restrict__ Wout, const float* __restrict__ bout,
             float* __restrict__ out) {
    int t = blockIdx.x * blockDim.x + threadIdx.x;
    if (t >= GN_GRAPHS * GN_DOUT) return;
    int g = t / GN_DOUT;
    int o = t % GN_DOUT;
    float inv = 1.0f / fmaxf(counts[g], 1.0f);
    float acc = bout[o];
    const float* sg = sums + (size_t)g * GN_D;
    #pragma unroll 4
    for (int k = 0; k < GN_D; ++k)
        acc += sg[k] * inv * Wout[(size_t)k * GN_DOUT + o];
    out[t] = acc;
}

// ---------------------------------------------------------------------------
extern "C" void kernel_launch(void* const* d_in, const int* in_sizes, int n_in,
                              void* d_out, int out_size, void* d_ws, size_t ws_size,
                              hipStream_t stream) {
    const float* x     = (const float*)d_in[0];
    const int*   eidx  = (const int*)d_in[1];     // (2, N_EDGES)
    const int*   batch = (const int*)d_in[2];
    const float* W1    = (const float*)d_in[3];   // (3,128,128)
    const float* b1    = (const float*)d_in[4];
    const float* gamma = (const float*)d_in[5];
    const float* beta  = (const float*)d_in[6];
    const float* rmean = (const float*)d_in[7];
    const float* rvar  = (const float*)d_in[8];
    const float* W2    = (const float*)d_in[9];
    const float* b2    = (const float*)d_in[10];
    const float* epsP  = (const float*)d_in[11];
    const float* Wout  = (const float*)d_in[12];
    const float* bout  = (const float*)d_in[13];
    float* out = (float*)d_out;

    const int* src = eidx;
    const int* dst = eidx + GN_EDGES;

    const size_t NF = (size_t)GN_NODES * GN_D;          // 3.2M floats
    float* bufP   = (float*)d_ws;
    float* bufQ   = bufP + NF;
    float* bufZ   = bufQ + NF;
    float* sums   = bufZ + NF;                          // 512*128
    float* counts = sums + (size_t)GN_GRAPHS * GN_D;    // 512

    const int nElem = (int)NF;
    const int gemmBlocks = (GN_MTILES + 3) / 4;         // 4 waves (M tiles) / block
    const int edgeThreads = GN_EDGES * 32;
    const int nodeThreads = GN_NODES * 32;

    const float* hin = x;
    for (int i = 0; i < GN_LAYERS; ++i) {
        float* agg = (i & 1) ? bufQ : bufP;
        gin_init_agg<<<(nElem + 255) / 256, 256, 0, stream>>>(hin, agg, epsP, i, nElem);
        gin_edge_agg<<<(edgeThreads + 255) / 256, 256, 0, stream>>>(hin, src, dst, agg, GN_EDGES);
        gin_gemm_epilogue<true><<<gemmBlocks, 128, 0, stream>>>(
            agg, W1 + (size_t)i * GN_D * GN_D, b1 + (size_t)i * GN_D,
            gamma + (size_t)i * GN_D, beta + (size_t)i * GN_D,
            rmean + (size_t)i * GN_D, rvar + (size_t)i * GN_D,
            bufZ, GN_NODES);
        // second linear: overwrite agg buffer (its data is dead now)
        gin_gemm_epilogue<false><<<gemmBlocks, 128, 0, stream>>>(
            bufZ, W2 + (size_t)i * GN_D * GN_D, b2 + (size_t)i * GN_D,
            nullptr, nullptr, nullptr, nullptr,
            agg, GN_NODES);
        hin = agg;
    }

    // readout
    const int poolN = GN_GRAPHS * GN_D + GN_GRAPHS;     // sums + counts contiguous
    gin_zero<<<(poolN + 255) / 256, 256, 0, stream>>>(sums, poolN);
    gin_pool<<<(nodeThreads + 255) / 256, 256, 0, stream>>>(hin, batch, sums, counts, GN_NODES);
    gin_out<<<(GN_GRAPHS * GN_DOUT + 255) / 256, 256, 0, stream>>>(sums, counts, Wout, bout, out);
}